// GGNNMsgPass_69930657513886
// MI455X (gfx1250) — compile-verified
//
#include <hip/hip_runtime.h>
#include <hip/hip_bf16.h>

typedef __attribute__((ext_vector_type(2))) float v2f;
typedef __attribute__((ext_vector_type(4))) float v4f;
typedef __attribute__((ext_vector_type(8))) float v8f;

#define BATCH 256
#define N 128
#define NF 128
#define NE 6               // edge types incl. "no edge" 0
#define TBL (NE * NE * NF) // 4608 floats

// ---------------------------------------------------------------------------
// Kernel 1: M[a][e][m] = sum_n W_ext[e,m,n] * A_ext[a,n]   (6 x 6 x 128 table)
// ---------------------------------------------------------------------------
__global__ __launch_bounds__(256) void ggnn_build_table(
    const float* __restrict__ adj_w,   // (5,128,128) [e-1][m][n]
    const float* __restrict__ adj_a,   // (5,128)     [a-1][n]
    float* __restrict__ Mtab)          // (6,6,128)   [a][e][m]
{
  const int t = threadIdx.x;
  for (int p = t; p < TBL; p += 256) {
    const int m = p & (NF - 1);
    const int e = (p >> 7) % NE;
    const int a = p / (NE * NF);
    float s = 0.0f;
    if (a > 0 && e > 0) {
      const float* wr = adj_w + ((e - 1) * NF + m) * NF;
      const float* ar = adj_a + (a - 1) * NF;
      #pragma unroll 8
      for (int n = 0; n < NF; ++n) s += wr[n] * ar[n];
    }
    Mtab[p] = s;
  }
}

// ---------------------------------------------------------------------------
// Kernel 2: one block per batch b.
//   cnt[j,e] = #{ i : bfm[b,i,j]==e }
//   msg[j,m] = sum_e cnt[j,e] * M[a_bfm[b,j]][e][m]
//   out[b]   = adj[b] @ msg          (f32 WMMA 16x16x4 GEMM)
// ---------------------------------------------------------------------------
#define ADJ_S 36    // 32 + 4 pad: A b64 column reads conflict-free
#define MSG_P 288   // dwords per K-pair row; 288 % 64 == 32 -> halves disjoint

__global__ __launch_bounds__(256) void ggnn_msgpass_fused(
    const int* __restrict__ bfm,     // (256,128,128) [b][i][j]
    const int* __restrict__ a_bfm,   // (256,128)     [b][j]
    const float* __restrict__ adj,   // (256,128,128) [b][i][j]
    const float* __restrict__ Mtab,  // (6,6,128)
    float* __restrict__ out)         // (256,128,128) [b][i][m]
{
  const int b    = blockIdx.x;
  const int t    = threadIdx.x;
  const int lane = t & 31;
  const int w    = t >> 5;        // wave id 0..7 -> output row stripe
  const int half = lane >> 4;     // half-wave for WMMA fragment layout
  const int l16  = lane & 15;

  __shared__ float sM[TBL];            // 18432 B
  __shared__ int   sCnt[32 * 6];       //   768 B
  __shared__ float sAdj[128 * ADJ_S];  // 18432 B
  // msg chunk, K-pair interleaved: (k,n) at (k>>1)*MSG_P + 2n + (k&1)
  __shared__ float sMsg[16 * MSG_P];   // 18432 B

  // table -> LDS as b128
  for (int p = t; p < TBL / 4; p += 256)
    reinterpret_cast<v4f*>(sM)[p] = reinterpret_cast<const v4f*>(Mtab)[p];

  const int bbase = b * (N * N);

  v8f acc[8] = {};   // 8 column-tiles of this wave's 16-row stripe

  for (int jc = 0; jc < 4; ++jc) {
    __syncthreads();  // previous chunk's WMMA reads finished

    // --- zero counters + stage adj[:, chunk] into LDS as b128 --------------
    if (t < 192) sCnt[t] = 0;
    #pragma unroll
    for (int k = 0; k < 4; ++k) {
      const int p = k * 256 + t;          // 1024 float4 = 128 rows x 8
      const int row = p >> 3, c4 = (p & 7) * 4;
      const v4f v = *reinterpret_cast<const v4f*>(
          &adj[bbase + row * N + jc * 32 + c4]);
      *reinterpret_cast<v4f*>(&sAdj[row * ADJ_S + c4]) = v;
    }
    __syncthreads();

    // --- histogram bfm[b, :, chunk] over edge types ------------------------
    {
      const int jl = t & 31, part = t >> 5;   // 8 parts x 16 rows of i
      const int* src = bfm + bbase + part * 16 * N + jc * 32 + jl;
      int c1 = 0, c2 = 0, c3 = 0, c4 = 0, c5 = 0;
      #pragma unroll
      for (int ii = 0; ii < 16; ++ii) {
        const int v = src[ii * N];
        c1 += (v == 1); c2 += (v == 2); c3 += (v == 3);
        c4 += (v == 4); c5 += (v == 5);
      }
      if (c1) atomicAdd(&sCnt[jl * 6 + 1], c1);
      if (c2) atomicAdd(&sCnt[jl * 6 + 2], c2);
      if (c3) atomicAdd(&sCnt[jl * 6 + 3], c3);
      if (c4) atomicAdd(&sCnt[jl * 6 + 4], c4);
      if (c5) atomicAdd(&sCnt[jl * 6 + 5], c5);
    }
    __syncthreads();

    // --- messages chunk: msg[j][m] = sum_e cnt[j][e] * M[a][e][m] ----------
    for (int k = 0; k < 16; ++k) {
      const int p = k * 256 + t;
      const int m = p & (NF - 1);
      const int j = p >> 7;                     // 0..31 (uniform per wave)
      const int a = a_bfm[b * N + jc * 32 + j];
      const float* Mr = &sM[a * (NE * NF) + m];
      const float s =
          (float)sCnt[j * 6 + 1] * Mr[1 * NF] +
          (float)sCnt[j * 6 + 2] * Mr[2 * NF] +
          (float)sCnt[j * 6 + 3] * Mr[3 * NF] +
          (float)sCnt[j * 6 + 4] * Mr[4 * NF] +
          (float)sCnt[j * 6 + 5] * Mr[5 * NF];
      sMsg[(j >> 1) * MSG_P + (m << 1) + (j & 1)] = s;
    }
    __syncthreads();

    // --- GEMM accumulate: acc += adj_chunk (128x32) @ msg_chunk (32x128) ---
    // A 16x4 f32 frag: v0 = K0|K2, v1 = K1|K3 (half-waves split K by 2).
    #pragma unroll
    for (int k4 = 0; k4 < 8; ++k4) {
      const int kb = k4 * 4 + half * 2;         // even
      const v2f A = *reinterpret_cast<const v2f*>(
          &sAdj[(w * 16 + l16) * ADJ_S + kb]);  // single ds_load_b64
      #pragma unroll
      for (int tm = 0; tm < 8; ++tm) {
        const v2f B = *reinterpret_cast<const v2f*>(
            &sMsg[(kb >> 1) * MSG_P + ((tm * 16 + l16) << 1)]); // ds_load_b64
        acc[tm] = __builtin_amdgcn_wmma_f32_16x16x4_f32(
            /*neg_a=*/false, A, /*neg_b=*/false, B,
            /*c_mod=*/(short)0, acc[tm],
            /*reuse_a=*/false, /*reuse_b=*/false);
      }
    }
  }

  // --- store D tiles: VGPR r -> row w*16 + r + 8*half, col tm*16 + l16 -----
  float* outb = out + bbase;
  #pragma unroll
  for (int tm = 0; tm < 8; ++tm) {
    #pragma unroll
    for (int r = 0; r < 8; ++r) {
      outb[(w * 16 + r + half * 8) * NF + tm * 16 + l16] = acc[tm][r];
    }
  }
}

// ---------------------------------------------------------------------------
extern "C" void kernel_launch(void* const* d_in, const int* in_sizes, int n_in,
                              void* d_out, int out_size, void* d_ws, size_t ws_size,
                              hipStream_t stream) {
  // setup_inputs order: afm(0, unused), bfm(1), a_bfm(2), adj(3), adj_w(4), adj_a(5)
  const int*   bfm   = (const int*)d_in[1];
  const int*   a_bfm = (const int*)d_in[2];
  const float* adj   = (const float*)d_in[3];
  const float* adj_w = (const float*)d_in[4];
  const float* adj_a = (const float*)d_in[5];
  float* Mtab = (float*)d_ws;          // 4608 floats (18 KB) of scratch
  float* outp = (float*)d_out;

  ggnn_build_table<<<1, 256, 0, stream>>>(adj_w, adj_a, Mtab);
  ggnn_msgpass_fused<<<BATCH, 256, 0, stream>>>(bfm, a_bfm, adj, Mtab, outp);
}